// AttributeDecoder_6760278524059
// MI455X (gfx1250) — compile-verified
//
#include <hip/hip_runtime.h>

typedef __attribute__((ext_vector_type(2))) float v2f;
typedef __attribute__((ext_vector_type(8))) float v8f;

#define NODES 100000
#define FIN   64
#define FMID  128
#define FOUT  256

// ---------------------------------------------------------------------------
// Degree / normalization kernels
// ---------------------------------------------------------------------------
__global__ void gcn2_deg_init(float* __restrict__ deg, int n) {
    int i = blockIdx.x * blockDim.x + threadIdx.x;
    if (i < n) deg[i] = 1.0f;           // self-loop contributes 1
}

__global__ void gcn2_deg_count(const int* __restrict__ col, float* __restrict__ deg,
                               long long E) {
    long long e = (long long)blockIdx.x * blockDim.x + threadIdx.x;
    if (e < E) atomicAdd(&deg[col[e]], 1.0f);
}

__global__ void gcn2_rsqrt(float* __restrict__ dis, int n) {
    int i = blockIdx.x * blockDim.x + threadIdx.x;
    if (i < n) dis[i] = rsqrtf(dis[i]); // deg >= 1 always
}

// ---------------------------------------------------------------------------
// Self-loop init: agg[i] = src[i] * dis[i]^2   (writes, no atomics)
// ---------------------------------------------------------------------------
template <int F>
__global__ void gcn2_self(const float* __restrict__ src, float* __restrict__ dst,
                          const float* __restrict__ dis, int n) {
    long long t = (long long)blockIdx.x * blockDim.x + threadIdx.x;
    if (t >= (long long)n * F) return;
    int i = (int)(t / F);
    float d = dis[i];
    dst[t] = src[t] * d * d;
}

// ---------------------------------------------------------------------------
// Edge scatter: agg[col] += src[row] * dis[row]*dis[col]
// F/4 threads per edge, float4 gather + 4 global f32 atomics (L2 resident)
// ---------------------------------------------------------------------------
template <int F>
__global__ void gcn2_scatter(const float* __restrict__ src, float* __restrict__ dst,
                             const int* __restrict__ row, const int* __restrict__ col,
                             const float* __restrict__ dis, long long E) {
    constexpr int TPE = F / 4;
    long long t = (long long)blockIdx.x * blockDim.x + threadIdx.x;
    long long e = t / TPE;
    if (e >= E) return;
    int c  = (int)(t % TPE) * 4;
    int r  = row[e];
    int cl = col[e];
    float nrm = dis[r] * dis[cl];
    const float4 v = *(const float4*)(src + (size_t)r * F + c);
    float* d = dst + (size_t)cl * F + c;
    atomicAdd(d + 0, v.x * nrm);
    atomicAdd(d + 1, v.y * nrm);
    atomicAdd(d + 2, v.z * nrm);
    atomicAdd(d + 3, v.w * nrm);
}

// ---------------------------------------------------------------------------
// fp32 WMMA GEMM: C[M x N] = A[M x K] @ B[K x N] + bias (optional relu)
// One wave computes a 16 x (16*NT) output strip using V_WMMA_F32_16X16X4_F32.
// A fragment is reused across NT independent WMMAs (pipeline friendly).
// M must be a multiple of 16 (100000 = 6250*16).
// ---------------------------------------------------------------------------
template <int K, int N, bool RELU>
__global__ void gcn2_gemm_wmma(const float* __restrict__ A, const float* __restrict__ B,
                               const float* __restrict__ bias, float* __restrict__ C,
                               int M) {
    constexpr int NT   = 4;            // 16-wide N tiles per wave
    constexpr int NBLK = N / (16 * NT);
    int wave  = (blockIdx.x * blockDim.x + threadIdx.x) >> 5;
    int lane  = threadIdx.x & 31;
    int mTile = wave / NBLK;
    int nBlk  = wave % NBLK;
    if (mTile * 16 >= M) return;       // wave-uniform: EXEC stays all-ones

    const int q     = lane & 15;       // position within 16-lane half
    const int half  = lane >> 4;       // 0: K pair (k0,k0+1) / rows 0-7 ; 1: (k0+2,k0+3) / rows 8-15
    const int mBase = mTile * 16;
    const int nBase = nBlk * 16 * NT;

    v8f acc[NT] = {};                  // 16x16 f32 accumulators

    const float* Arow = A + (size_t)(mBase + q) * K;
    for (int k0 = 0; k0 < K; k0 += 4) {
        const int kk = k0 + 2 * half;
        v2f a;
        a.x = Arow[kk];
        a.y = Arow[kk + 1];
#pragma unroll
        for (int t = 0; t < NT; ++t) {
            const int cIdx = nBase + t * 16 + q;
            v2f b;
            b.x = B[(size_t)kk * N + cIdx];
            b.y = B[(size_t)(kk + 1) * N + cIdx];
            acc[t] = __builtin_amdgcn_wmma_f32_16x16x4_f32(
                /*neg_a=*/false, a, /*neg_b=*/false, b,
                /*c_mod=*/(short)0, acc[t], /*reuse_a=*/false, /*reuse_b=*/false);
        }
    }

#pragma unroll
    for (int t = 0; t < NT; ++t) {
        const int cIdx = nBase + t * 16 + q;
        const float bv = bias[cIdx];
#pragma unroll
        for (int i = 0; i < 8; ++i) {
            const int m = mBase + 8 * half + i;   // C/D layout: row = vgpr + 8*(lane>=16)
            float v = acc[t][i] + bv;
            if (RELU) v = fmaxf(v, 0.0f);
            C[(size_t)m * N + cIdx] = v;
        }
    }
}

// ---------------------------------------------------------------------------
extern "C" void kernel_launch(void* const* d_in, const int* in_sizes, int n_in,
                              void* d_out, int out_size, void* d_ws, size_t ws_size,
                              hipStream_t stream) {
    const float* x   = (const float*)d_in[0];
    const int*   ei  = (const int*)d_in[1];
    const float* W1  = (const float*)d_in[2];
    const float* b1  = (const float*)d_in[3];
    const float* W2  = (const float*)d_in[4];
    const float* b2  = (const float*)d_in[5];

    const int       n = in_sizes[0] / FIN;       // 100000
    const long long E = in_sizes[1] / 2;         // 1600000
    const int* row = ei;                         // edge_index[0]
    const int* col = ei + E;                     // edge_index[1]

    // Workspace layout: [dis: n floats][agg: n*FMID floats]
    // agg1 (n*FIN) lives in the front of the agg slot; agg2 (n*FMID) reuses it.
    char* ws = (char*)d_ws;
    float* dis = (float*)ws;
    size_t disBytes = (((size_t)n * 4) + 255) & ~(size_t)255;
    float* agg = (float*)(ws + disBytes);

    float* h   = (float*)d_out;                  // stage h in d_out (dead before GEMM2 writes)
    float* out = (float*)d_out;

    const int B = 256;
    auto blocks = [](long long work, int bs) { return (int)((work + bs - 1) / bs); };

    // 1) degrees -> dis = rsqrt(deg)
    gcn2_deg_init<<<blocks(n, B), B, 0, stream>>>(dis, n);
    gcn2_deg_count<<<blocks(E, B), B, 0, stream>>>(col, dis, E);
    gcn2_rsqrt<<<blocks(n, B), B, 0, stream>>>(dis, n);

    // 2) layer 1: agg1 = Agg(x)  (aggregate BEFORE linear: Agg(x)@W1 == Agg(x@W1))
    gcn2_self<FIN><<<blocks((long long)n * FIN, B), B, 0, stream>>>(x, agg, dis, n);
    gcn2_scatter<FIN><<<blocks(E * (FIN / 4), B), B, 0, stream>>>(x, agg, row, col, dis, E);

    // 3) h = relu(agg1 @ W1 + b1)   [100000x64]@[64x128]
    {
        const int waves = (n / 16) * (FMID / 64);
        gcn2_gemm_wmma<FIN, FMID, true>
            <<<blocks((long long)waves * 32, B), B, 0, stream>>>(agg, W1, b1, h, n);
    }

    // 4) layer 2: agg2 = Agg(h)
    gcn2_self<FMID><<<blocks((long long)n * FMID, B), B, 0, stream>>>(h, agg, dis, n);
    gcn2_scatter<FMID><<<blocks(E * (FMID / 4), B), B, 0, stream>>>(h, agg, row, col, dis, E);

    // 5) out = agg2 @ W2 + b2      [100000x128]@[128x256]
    {
        const int waves = (n / 16) * (FOUT / 64);
        gcn2_gemm_wmma<FMID, FOUT, false>
            <<<blocks((long long)waves * 32, B), B, 0, stream>>>(agg, W2, b2, out, n);
    }
}